// CentroidDistance_11536282157451
// MI455X (gfx1250) — compile-verified
//
#include <hip/hip_runtime.h>
#include <cstdint>

typedef float v2f __attribute__((ext_vector_type(2)));
typedef float v8f __attribute__((ext_vector_type(8)));

#define PB_EPS 1e-6f

static constexpr int Bsz  = 32;    // batches
static constexpr int Npts = 4096;  // points per batch
static constexpr int Edim = 64;    // embedding dim (K)
static constexpr int Cc   = 256;   // centroids
static constexpr int LDSTR = 68;   // padded centroid row stride in floats (272B: 16B-aligned, +4 bank rotation/row)

// ---------------------------------------------------------------------------
// Main kernel: one block = 128 rows (n) of one batch b, against all 256 centroids.
// 8 waves x (16 rows each) x 16 c-tiles of 16, K=64 via 16 chained
// V_WMMA_F32_16X16X4_F32. Writes dist, emits deterministic per-block column sums.
// ---------------------------------------------------------------------------
__global__ __launch_bounds__(256) void poincare_main_kernel(
    const float* __restrict__ x,     // (B, N, E)
    const float* __restrict__ mask,  // (B, N, 1)
    const float* __restrict__ cent,  // (C, E)
    float* __restrict__ out,         // [0 : B*C) graph_dist, [B*C : ...) dist (B,N,C)
    float* __restrict__ part)        // (B, 32 strips, C) partial column sums
{
  __shared__ float sC[Cc * LDSTR];   // centroid tile, padded rows
  __shared__ float sC2[Cc];          // ||c||^2
  __shared__ float sPart[8 * Cc];    // per-wave column sums

  const int tid   = threadIdx.x;
  const int b     = blockIdx.y;
  const int strip = blockIdx.x;      // 0..31 -> rows [strip*128, strip*128+128)
  const int wave  = tid >> 5;
  const int lane  = tid & 31;
  const int lr    = lane & 15;       // M (A) / N (B,C/D) index within tile
  const int khalf = lane >> 4;       // selects K-pair within a fragment

  // ---- stage centroids into LDS (float4 global loads, padded LDS rows)
  for (int i = tid; i < Cc * (Edim / 4); i += 256) {
    const int row = i >> 4, q = i & 15;
    const float4 v = reinterpret_cast<const float4*>(cent)[row * (Edim / 4) + q];
    float* d = &sC[row * LDSTR + q * 4];
    d[0] = v.x; d[1] = v.y; d[2] = v.z; d[3] = v.w;
  }
  __syncthreads();

  { // ||c||^2 per centroid (one thread per centroid row)
    float s = 0.f;
    const float* r = &sC[tid * LDSTR];
    #pragma unroll
    for (int e = 0; e < Edim; ++e) s += r[e] * r[e];
    sC2[tid] = s;
  }
  __syncthreads();

  // ---- A fragments: 16 rows x K=64. ISA layout for 16x4 f32 A:
  // lane L, VGPR v -> A[M = L%16][K = 4k + 2*(L/16) + v]
  const int n0 = strip * 128 + wave * 16;
  const float* xrow = x + ((size_t)b * Npts + n0 + lr) * Edim + khalf * 2;
  v2f a[16];
  #pragma unroll
  for (int k = 0; k < 16; ++k)
    a[k] = *reinterpret_cast<const v2f*>(xrow + 4 * k);

  // ||x||^2 per row: each lane holds half the row's K values; fold halves.
  float s2 = 0.f;
  #pragma unroll
  for (int k = 0; k < 16; ++k) s2 += a[k].x * a[k].x + a[k].y * a[k].y;
  s2 += __shfl_xor(s2, 16);  // lane L now holds ||x_{n0+L%16}||^2

  // Gather x2 + mask for the 8 C/D rows this lane's accumulators cover
  // (C/D layout: VGPR v, lane L -> M = v + 8*(L/16)).
  float x2v[8], mkv[8];
  #pragma unroll
  for (int v = 0; v < 8; ++v) {
    x2v[v] = __shfl(s2, khalf * 8 + v);
    mkv[v] = mask[(size_t)b * Npts + n0 + khalf * 8 + v];
  }

  float* dist = out + (size_t)Bsz * Cc;

  for (int ct = 0; ct < 16; ++ct) {
    v8f acc = {};
    // B layout (4x16 f32): lane L, VGPR v -> B[K = 4k + 2*(L/16) + v][N = L%16]
    const float* bbase = &sC[(ct * 16 + lr) * LDSTR + khalf * 2];
    #pragma unroll
    for (int k = 0; k < 16; ++k) {
      v2f bf = *reinterpret_cast<const v2f*>(bbase + 4 * k);
      acc = __builtin_amdgcn_wmma_f32_16x16x4_f32(
          /*neg_a=*/false, a[k], /*neg_b=*/false, bf,
          /*c_mod=*/(short)0, acc, /*reuse_a=*/false, /*reuse_b=*/false);
    }

    const float c2  = sC2[ct * 16 + lr];
    const float omc = 1.f - c2;
    float colsum = 0.f;
    #pragma unroll
    for (int v = 0; v < 8; ++v) {
      const float x2 = x2v[v];
      float sq  = fmaxf(x2 + c2 - 2.f * acc[v], 0.f);      // ||x-c||^2
      float den = (1.f - x2) * omc;
      float t   = 2.f * sq / fmaxf(den, PB_EPS);           // arg - 1
      t = fmaxf(t, PB_EPS);
      // arccosh(1+t) = log1p(t + sqrt(t*(t+2))) -- stable near t=0
      float d = log1pf(t + sqrtf(t * (t + 2.f)));
      d *= mkv[v];
      dist[((size_t)(b * Npts + n0 + khalf * 8 + v)) * Cc + ct * 16 + lr] = d;
      colsum += d;
    }
    colsum += __shfl_xor(colsum, 16);                      // sum over 16 rows of tile
    if (lane < 16) sPart[wave * Cc + ct * 16 + lr] = colsum;
  }
  __syncthreads();

  // Fixed-order cross-wave fold -> deterministic per-block partial
  float bsum = 0.f;
  #pragma unroll
  for (int w = 0; w < 8; ++w) bsum += sPart[w * Cc + tid];
  part[((size_t)b * 32 + strip) * Cc + tid] = bsum;
}

// ---------------------------------------------------------------------------
// Finalize: graph_dist[b][c] = sum_strip part[b][strip][c] / sum_n mask[b][n]
// (fixed reduction order -> deterministic)
// ---------------------------------------------------------------------------
__global__ __launch_bounds__(256) void poincare_finalize_kernel(
    const float* __restrict__ mask, const float* __restrict__ part,
    float* __restrict__ out)
{
  __shared__ float red[256];
  const int b = blockIdx.x, tid = threadIdx.x;

  float s = 0.f;
  for (int n = tid; n < Npts; n += 256) s += mask[(size_t)b * Npts + n];
  red[tid] = s;
  __syncthreads();
  for (int off = 128; off > 0; off >>= 1) {
    if (tid < off) red[tid] += red[tid + off];
    __syncthreads();
  }
  const float inv = 1.f / red[0];

  float g = 0.f;
  #pragma unroll
  for (int st = 0; st < 32; ++st)
    g += part[((size_t)b * 32 + st) * Cc + tid];
  out[b * Cc + tid] = g * inv;
}

extern "C" void kernel_launch(void* const* d_in, const int* in_sizes, int n_in,
                              void* d_out, int out_size, void* d_ws, size_t ws_size,
                              hipStream_t stream) {
  (void)in_sizes; (void)n_in; (void)out_size; (void)ws_size;
  const float* x    = (const float*)d_in[0];  // node_repr (32,4096,64)
  const float* mask = (const float*)d_in[1];  // mask      (32,4096,1)
  const float* cent = (const float*)d_in[2];  // centroid  (256,64)
  float* out  = (float*)d_out;                // [graph_dist | dist]
  float* part = (float*)d_ws;                 // 32*32*256 floats = 1 MiB scratch

  dim3 grid(32, 32);  // (strips, batches)
  poincare_main_kernel<<<grid, 256, 0, stream>>>(x, mask, cent, out, part);
  poincare_finalize_kernel<<<32, 256, 0, stream>>>(mask, part, out);
}